// ModelNew_25056839205320
// MI455X (gfx1250) — compile-verified
//
#include <hip/hip_runtime.h>
#include <hip/hip_bf16.h>

typedef __attribute__((ext_vector_type(2))) float v2f;
typedef __attribute__((ext_vector_type(4))) float v4f;
typedef __attribute__((ext_vector_type(8))) float v8f;

#define EPS 1e-5f
#define N_ 64
#define C_ 256
#define HW_ 3136              // 56*56
#define ROWS_ (N_ * C_)       // 16384
#define F4_PER_ROW (HW_ / 4)  // 784
#define TOTAL4 ((unsigned)ROWS_ * (unsigned)F4_PER_ROW)  // 12,845,056 = 50176*256

// ---------------------------------------------------------------------------
// Kernel 1: per-(n,c) sum of squares over HW. One wave32 per row.
// Main loop: float4 loads (512B per wave per iter), VALU x^2 accumulation.
// Final cross-lane reduction: ONE v_wmma_f32_16x16x4_f32 with A = ones folds
// the 64 per-lane partials (B matrix) into replicated column sums, then a
// short shuffle tree finishes: sum_lanes(acc[0]) == 2 * total (exact fp32).
// ---------------------------------------------------------------------------
__global__ __launch_bounds__(256) void k_rowsumsq(const float* __restrict__ x,
                                                  float* __restrict__ sumsq) {
  const int wave = threadIdx.x >> 5;
  const int lane = threadIdx.x & 31;
  const int row  = blockIdx.x * 8 + wave;          // 0..16383, always valid

  const v4f* __restrict__ xr =
      reinterpret_cast<const v4f*>(x + (size_t)row * HW_);

  v4f a; a.x = 0.f; a.y = 0.f; a.z = 0.f; a.w = 0.f;
  // 24 iterations * 128 floats = 3072 floats
  #pragma unroll 4
  for (int it = 0; it < 24; ++it) {
    __builtin_prefetch(&xr[(it + 6) * 32 + lane], 0, 0);  // global_prefetch (speculative)
    v4f v = xr[it * 32 + lane];
    a.x += v.x * v.x; a.y += v.y * v.y;
    a.z += v.z * v.z; a.w += v.w * v.w;
  }
  // tail: 64 floats (3072..3135), float2 per lane
  const v2f* __restrict__ xt =
      reinterpret_cast<const v2f*>(x + (size_t)row * HW_ + 3072);
  v2f t = xt[lane];

  v2f b;                                  // 2 fp32 partials/lane = B matrix 4x16
  b.x = a.x + a.y + t.x * t.x;
  b.y = a.z + a.w + t.y * t.y;

  v2f ones; ones.x = 1.0f; ones.y = 1.0f; // A matrix 16x4 = all ones
  v8f acc = {};
  // D[m][n] = sum_k B[k][n]  (A==1). Every D element is a column sum.
  acc = __builtin_amdgcn_wmma_f32_16x16x4_f32(
      /*neg_a=*/false, ones, /*neg_b=*/false, b,
      /*c_mod=*/(short)0, acc, /*reuse_a=*/false, /*reuse_b=*/false);

  // sum acc[0] over 32 lanes = row0 colsums (lanes 0-15) + row8 colsums (16-31)
  //                          = 2 * total sum of squares
  float r = acc[0];
  #pragma unroll
  for (int off = 16; off >= 1; off >>= 1) r += __shfl_xor(r, off, 32);
  if (lane == 0) sumsq[row] = 0.5f * r;
}

// ---------------------------------------------------------------------------
// Kernel 2: per-n gate. One block (256 threads) per n; thread == channel c.
// embed = sqrt(sumsq+eps)*alpha; inv = rsqrt(mean_c(embed^2)+eps);
// gate = 1 + tanh(embed*gamma*inv + beta).   16384 outputs total — tiny.
// ---------------------------------------------------------------------------
__global__ __launch_bounds__(256) void k_gate(const float* __restrict__ sumsq,
                                              const float* __restrict__ alpha,
                                              const float* __restrict__ gammav,
                                              const float* __restrict__ betav,
                                              float* __restrict__ gate) {
  const int n = blockIdx.x;
  const int c = threadIdx.x;

  float e  = sqrtf(sumsq[n * C_ + c] + EPS) * alpha[c];
  float r  = e * e;
  #pragma unroll
  for (int off = 16; off >= 1; off >>= 1) r += __shfl_xor(r, off, 32);

  __shared__ float red[8];
  if ((threadIdx.x & 31) == 0) red[threadIdx.x >> 5] = r;
  __syncthreads();
  float s = red[0] + red[1] + red[2] + red[3] +
            red[4] + red[5] + red[6] + red[7];

  float inv = rsqrtf(s * (1.0f / (float)C_) + EPS);
  float z   = e * gammav[c] * inv + betav[c];
  gate[n * C_ + c] = 1.0f + tanhf(z);
}

// ---------------------------------------------------------------------------
// Kernel 3: out = x * gate[n,c], float4 streaming. Traverse x in REVERSE so
// the tail of x still resident in the 192MB L2 (from kernel 1's forward
// stream) is consumed before eviction. Non-temporal stores for out (never
// re-read) keep x's lines in L2.
// ---------------------------------------------------------------------------
__global__ __launch_bounds__(256) void k_scale(const float* __restrict__ x,
                                               const float* __restrict__ gate,
                                               float* __restrict__ out) {
  unsigned flat = blockIdx.x * 256u + threadIdx.x;   // < TOTAL4
  unsigned j    = (TOTAL4 - 1u) - flat;              // reverse traversal
  unsigned nc   = j / (unsigned)F4_PER_ROW;          // magic-mul division

  float g = gate[nc];
  v4f v = reinterpret_cast<const v4f*>(x)[j];
  v4f o;
  o.x = v.x * g; o.y = v.y * g; o.z = v.z * g; o.w = v.w * g;
  __builtin_nontemporal_store(o, reinterpret_cast<v4f*>(out) + j);
}

// ---------------------------------------------------------------------------
extern "C" void kernel_launch(void* const* d_in, const int* in_sizes, int n_in,
                              void* d_out, int out_size, void* d_ws, size_t ws_size,
                              hipStream_t stream) {
  const float* x      = (const float*)d_in[0];
  const float* alpha  = (const float*)d_in[1];
  const float* gammav = (const float*)d_in[2];
  const float* betav  = (const float*)d_in[3];
  float* out = (float*)d_out;

  float* sumsq = (float*)d_ws;          // 16384 floats
  float* gate  = sumsq + ROWS_;         // 16384 floats (128 KB total scratch)

  k_rowsumsq<<<ROWS_ / 8, 256, 0, stream>>>(x, sumsq);
  k_gate<<<N_, 256, 0, stream>>>(sumsq, alpha, gammav, betav, gate);
  k_scale<<<TOTAL4 / 256, 256, 0, stream>>>(x, gate, out);
}